// _CustomNextItemPredictionTask_42640435315363
// MI455X (gfx1250) — compile-verified
//
#include <hip/hip_runtime.h>

typedef float v2f __attribute__((ext_vector_type(2)));
typedef float v4f __attribute__((ext_vector_type(4)));
typedef float v8f __attribute__((ext_vector_type(8)));

#define BS    4096
#define DDIM  128
#define SEQ   20
#define NNEG  16384
#define UU    (BS + NNEG)   // 20480
#define SORTN 32768

// ---------------- sort setup + bitonic step (global memory) ----------------

__global__ void fill_sort(const int* __restrict__ labels, const int* __restrict__ uneg,
                          int* __restrict__ buf) {
  int i = blockIdx.x * blockDim.x + threadIdx.x;
  if (i >= SORTN) return;
  int v;
  if (i < BS)      v = labels[i];
  else if (i < UU) v = uneg[i - BS];
  else             v = 0x7FFFFFFF;
  buf[i] = v;
}

__global__ void bitonic_step(int* __restrict__ a, int j, int k) {
  int i = blockIdx.x * blockDim.x + threadIdx.x;
  int ixj = i ^ j;
  if (ixj > i) {
    int x = a[i], y = a[ixj];
    bool up = ((i & k) == 0);
    if (up ? (x > y) : (x < y)) { a[i] = y; a[ixj] = x; }
  }
}

// ---------------- gather embedding rows + per-column log bias ----------------

__global__ void gather_prep(const int* __restrict__ uniq, const float* __restrict__ emb,
                            const float* __restrict__ probs, float* __restrict__ G,
                            float* __restrict__ logp) {
  int g = blockIdx.x * blockDim.x + threadIdx.x;
  int u = g >> 5;          // row of G
  int c = g & 31;          // float4 chunk within row (32 * 4 = 128 floats)
  if (u >= UU) return;
  int lbl = uniq[u];
  // emb rows are touched exactly once here -> non-temporal load
  const v4f* src = (const v4f*)(emb + (size_t)lbl * DDIM);
  v4f val = __builtin_nontemporal_load(src + c);
  ((v4f*)(G + (size_t)u * DDIM))[c] = val;        // G is re-read by GEMM: keep RT
  if (c == 0) logp[u] = logf(probs[lbl] + 1e-11f);
}

// ---------------- positives via binary search + zero second output ----------------

__device__ __forceinline__ int bsearch_eq(const int* __restrict__ a, int n, int key) {
  int lo = 0, hi = n - 1;
  while (lo <= hi) {
    int mid = (lo + hi) >> 1;
    int v = a[mid];
    if (v == key) return mid;
    if (v < key) lo = mid + 1; else hi = mid - 1;
  }
  return -1;
}

__global__ void pos_zero(const int* __restrict__ uniq, const int* __restrict__ labels,
                         int* __restrict__ pos, float* __restrict__ out) {
  int i = blockIdx.x * blockDim.x + threadIdx.x;
  if (i >= BS) return;
  pos[i] = bsearch_eq(uniq, UU, labels[i]);
  out[(size_t)BS * UU + i] = 0.0f;   // zeros_like(labels_all)
}

// ---------------- fused FP32 WMMA GEMM + bias + permuted store ----------------
// wave tile: 16 (M) x 64 (N); block: 8 waves -> 32 (M) x 256 (N)

__global__ __launch_bounds__(256)
void gemm_epilogue(const float* __restrict__ x, const float* __restrict__ G,
                   const float* __restrict__ logp, const int* __restrict__ pos,
                   const float* __restrict__ tp, const float* __restrict__ icfp,
                   float* __restrict__ out) {
  const int tid  = threadIdx.x;
  const int lane = tid & 31;
  const int wave = tid >> 5;
  const int m0 = blockIdx.y * 32 + (wave & 1) * 16;
  const int n0 = blockIdx.x * 256 + (wave >> 1) * 64;

  const int l15   = lane & 15;
  const int khalf = (lane >> 4) * 2;   // lanes 16..31 take K = k0+2, k0+3

  // A 16x4 f32 layout: lane<16 row M=l15 K={k0,k0+1}; lane>=16 same rows K={k0+2,k0+3}
  const float* ap  = x + (size_t)(m0 + l15) * DDIM + khalf;
  // B 4x16 f32 layout mirrors A with G row = column n
  const float* bp0 = G + (size_t)(n0 + l15) * DDIM + khalf;

  v8f acc[4] = {};

#pragma unroll
  for (int k0 = 0; k0 < DDIM; k0 += 4) {
    v2f a = *(const v2f*)(ap + k0);
#pragma unroll
    for (int t = 0; t < 4; ++t) {
      v2f b = *(const v2f*)(bp0 + (size_t)t * 16 * DDIM + k0);
      acc[t] = __builtin_amdgcn_wmma_f32_16x16x4_f32(
          /*neg_a=*/false, a, /*neg_b=*/false, b,
          /*c_mod=*/(short)0, acc[t], /*reuse_a=*/false, /*reuse_b=*/false);
    }
  }

  const float T   = *tp;
  const float icf = *icfp;
  float lb[4];
#pragma unroll
  for (int t = 0; t < 4; ++t) lb[t] = logp[n0 + t * 16 + l15];

  // C/D layout: VGPR v -> row m0 + v (+8 for lanes 16..31), col = n0t + (lane&15)
  const int rbase = m0 + ((lane >> 4) << 3);
#pragma unroll
  for (int v = 0; v < 8; ++v) {
    const int row = rbase + v;
    const int p = pos[row];
    float* orow = out + (size_t)row * UU;
#pragma unroll
    for (int t = 0; t < 4; ++t) {
      const int col = n0 + t * 16 + l15;
      float val = (acc[t][v] - icf * lb[t]) / T;
      int dest = (col == p) ? 0 : ((col < p) ? col + 1 : col);
      // 336 MB write-once stream: non-temporal so it doesn't sweep the 192 MB L2
      __builtin_nontemporal_store(val, orow + dest);
    }
  }
}

// ---------------- mask cleanup: zero history hits (<= 81920 elements) ----------------

__global__ void mask_cleanup(const int* __restrict__ uniq, const int* __restrict__ seq,
                             const int* __restrict__ pos, float* __restrict__ out) {
  int g = blockIdx.x * blockDim.x + threadIdx.x;
  if (g >= BS * SEQ) return;
  int i = g / SEQ;
  int u = bsearch_eq(uniq, UU, seq[g]);
  if (u < 0) return;
  int p = pos[i];
  if (u == p) return;                       // positives never masked
  int dest = (u < p) ? u + 1 : u;
  __builtin_nontemporal_store(0.0f, out + (size_t)i * UU + dest);
}

// ---------------- launch ----------------

extern "C" void kernel_launch(void* const* d_in, const int* in_sizes, int n_in,
                              void* d_out, int out_size, void* d_ws, size_t ws_size,
                              hipStream_t stream) {
  const float* x      = (const float*)d_in[0];
  const int*   labels = (const int*)d_in[1];
  const int*   seq    = (const int*)d_in[2];
  const float* probs  = (const float*)d_in[3];
  const int*   uneg   = (const int*)d_in[4];
  const float* emb    = (const float*)d_in[5];
  const float* temp   = (const float*)d_in[6];
  const float* icf    = (const float*)d_in[7];
  float* out = (float*)d_out;

  // workspace layout (~10.7 MB)
  int*   sortbuf = (int*)d_ws;                    // 32768 ints (sorted; first 20480 = unique_labels)
  float* logp    = (float*)(sortbuf + SORTN);     // 20480 floats
  int*   pos     = (int*)(logp + UU);             // 4096 ints
  float* G       = (float*)(pos + BS);            // 20480 x 128 floats (16B-aligned offset)

  fill_sort<<<SORTN / 256, 256, 0, stream>>>(labels, uneg, sortbuf);
  for (int k = 2; k <= SORTN; k <<= 1)
    for (int j = k >> 1; j > 0; j >>= 1)
      bitonic_step<<<SORTN / 256, 256, 0, stream>>>(sortbuf, j, k);

  gather_prep<<<(UU * 32) / 256, 256, 0, stream>>>(sortbuf, emb, probs, G, logp);
  pos_zero<<<BS / 256, 256, 0, stream>>>(sortbuf, labels, pos, out);

  dim3 grid(UU / 256, BS / 32);                   // 80 x 128 blocks, 256 threads
  gemm_epilogue<<<grid, 256, 0, stream>>>(x, G, logp, pos, temp, icf, out);

  mask_cleanup<<<(BS * SEQ + 255) / 256, 256, 0, stream>>>(sortbuf, seq, pos, out);
}